// SegmentRasteriser_6262062318106
// MI455X (gfx1250) — compile-verified
//
#include <hip/hip_runtime.h>

typedef __attribute__((ext_vector_type(2))) float v2f;
typedef __attribute__((ext_vector_type(8))) float v8f;

#define EPS_F 1.1920928955078125e-07f   // FLT_EPSILON, matches jnp.finfo(f32).eps

// One wave rasterises one 16x16 pixel tile of one (batch,segment) plane.
// t-field (affine in pixel coords => rank-2 16x16 matrix) is produced by a
// single V_WMMA_F32_16X16X4_F32; the clamp/distance/exp/colour tail is VALU.
// Output is a 201MB write-once stream (> 192MB L2), so stores are non-temporal.
__global__ __launch_bounds__(256) void seg_raster_kernel(
    const float* __restrict__ x,   // (4,64,8)
    float* __restrict__ out)       // (4,64,3,256,256)
{
    const int bs   = blockIdx.y;            // 0..255 : b*64 + s
    const int lane = threadIdx.x & 31;
    const int wave = threadIdx.x >> 5;      // 0..7
    const int tile = blockIdx.x * 8 + wave; // 0..255 tiles of the 256x256 plane
    const int ti0  = (tile >> 4) << 4;      // tile row origin (i)
    const int tj0  = (tile & 15) << 4;      // tile col origin (j)
    const int half = lane >> 4;             // 0: lanes 0-15, 1: lanes 16-31
    const int l16  = lane & 15;

    // ---- per-segment scalars (uniform address -> scalar loads) ----
    const float* xp = x + bs * 8;
    const float sx = xp[0] * 256.0f;
    const float sy = xp[1] * 256.0f;
    const float ex = xp[2] * 256.0f;
    const float ey = xp[3] * 256.0f;
    const float th = 1.0f + xp[4] * 127.0f;           // 1 + x4*(max(dims)/2 - 1)
    const float col0 = xp[5], col1 = xp[6], col2 = xp[7];

    const float mx = ex - sx, my = ey - sy;
    const float inv_mm = 1.0f / (mx * mx + my * my + EPS_F);
    const float fmx = mx * inv_mm;
    const float fmy = my * inv_mm;
    const float t0  = -(sx * fmx + sy * fmy);         // t(i,j)=fmx*i+fmy*j+t0
    const float sig = 0.54925f * th;
    const float cxp = -1.0f / (sig * sig + EPS_F);    // canvas = exp(d*d*cxp)

    // ---- rank-2 t-field via f32 WMMA (16x16x4) ----
    // A (16x4): lanes 0-15 hold K=0,1 for M=lane; lanes 16-31 hold K=2,3 (zero).
    // B (4x16): lanes 0-15 hold rows K=0,1 for N=lane; lanes 16-31 rows K=2,3 (zero).
    v2f A, B;
    A.x = half ? 0.0f : (fmx * (float)(ti0 + l16) + t0); // K=0 column of A
    A.y = half ? 0.0f : 1.0f;                            // K=1 column of A
    B.x = half ? 0.0f : 1.0f;                            // K=0 row of B
    B.y = half ? 0.0f : (fmy * (float)(tj0 + l16));      // K=1 row of B

    v8f acc = {};
    acc = __builtin_amdgcn_wmma_f32_16x16x4_f32(
        /*neg_a=*/false, A, /*neg_b=*/false, B,
        /*c_mod=*/(short)0, acc, /*reuse_a=*/false, /*reuse_b=*/false);

    // ---- per-pixel tail: clamp, squared distance, exp, 3-channel stores ----
    const float jy = (float)(tj0 + l16);
    // accumulator VGPR r holds row (ti0 + half*8 + r), column (tj0 + l16)
    float* p0 = out + (size_t)bs * 3u * 65536u
                    + (size_t)(ti0 + half * 8) * 256u + (size_t)(tj0 + l16);

#pragma unroll
    for (int r = 0; r < 8; ++r) {
        float t  = acc[r];
        float tc = fminf(fmaxf(t, 0.0f), 1.0f);      // clamp == 3-case select
        float ix = (float)(ti0 + half * 8 + r);
        float dx = (ix - sx) - tc * mx;              // ps - tc*m
        float dy = (jy - sy) - tc * my;
        float d  = fmaf(dx, dx, dy * dy);            // squared distance
        float v  = __expf(d * d * cxp);              // exp(-d^2/(sig^2+eps))
        __builtin_nontemporal_store(v * col0, p0 + r * 256);           // ch 0
        __builtin_nontemporal_store(v * col1, p0 + 65536 + r * 256);   // ch 1
        __builtin_nontemporal_store(v * col2, p0 + 131072 + r * 256);  // ch 2
    }
}

extern "C" void kernel_launch(void* const* d_in, const int* in_sizes, int n_in,
                              void* d_out, int out_size, void* d_ws, size_t ws_size,
                              hipStream_t stream) {
    (void)in_sizes; (void)n_in; (void)out_size; (void)d_ws; (void)ws_size;
    const float* x = (const float*)d_in[0];  // (4,64,8) fp32
    float* out = (float*)d_out;              // (4,64,3,256,256) fp32

    // 256 (b,s) planes on grid.y; 32 blocks * 8 waves = 256 tiles per plane.
    dim3 grid(32, 256, 1);
    dim3 block(256, 1, 1);
    hipLaunchKernelGGL(seg_raster_kernel, grid, block, 0, stream, x, out);
}